// BlockLinear_28836410426083
// MI455X (gfx1250) — compile-verified
//
#include <hip/hip_runtime.h>

typedef __attribute__((ext_vector_type(2))) float v2f;
typedef __attribute__((ext_vector_type(4))) float v4f;
typedef __attribute__((ext_vector_type(8))) float v8f;
typedef __attribute__((ext_vector_type(4))) int   v4i;

#define NTHREADS 128   // 4 waves (wave32)
#define TPW      8     // tokens per workgroup

// ---- CDNA5 async global->LDS path (guarded; falls back to sync loads) ----
#if defined(__AMDGCN__) && __has_builtin(__builtin_amdgcn_global_load_async_to_lds_b128) && \
    __has_builtin(__builtin_amdgcn_s_wait_asynccnt)
#define USE_ASYNC 1
#pragma message("CDNA5 async-to-LDS path ENABLED")
#else
#define USE_ASYNC 0
#pragma message("CDNA5 async-to-LDS path DISABLED (builtins missing) - sync fallback")
#endif

#if USE_ASYNC
typedef __attribute__((address_space(1))) v4i g_v4i;   // global int4
typedef __attribute__((address_space(3))) v4i l_v4i;   // LDS int4

// 8 x global_load_async_to_lds_b128 per wave per token (wave moves 4KB; WG 16KB).
__device__ __forceinline__ void async_copy_token(const float* __restrict__ g,
                                                 float* __restrict__ l, int tid) {
#pragma unroll
  for (int i = tid * 4; i < 4096; i += NTHREADS * 4) {
    g_v4i* gp = (g_v4i*)(unsigned long long)(g + i);
    l_v4i* lp = (l_v4i*)(unsigned long long)(unsigned int)(unsigned long long)(l + i);
    __builtin_amdgcn_global_load_async_to_lds_b128(gp, lp, 0, 0);
  }
}
#define WAIT_ASYNC(n) __builtin_amdgcn_s_wait_asynccnt(n)
#endif

// One GEMM step: src (LDS, row-major [32,128]) x W (LDS, row-major [128,128])
// -> dst (LDS), stored COLUMN-major (dst[n*32+m]) which realizes the
// transpose-reshape permutation of the reference for free.
__device__ __forceinline__ void gemm_perm_step(const float* __restrict__ src,
                                               const float* __restrict__ W,
                                               float* __restrict__ dst,
                                               int wave, int half, int l16) {
#pragma unroll
  for (int nt = 0; nt < 2; ++nt) {
    const int nbase = (wave * 2 + nt) * 16;
    const int n     = nbase + l16;          // this lane's N column
    v8f c0 = {};                            // M-tile 0 (rows 0..15)
    v8f c1 = {};                            // M-tile 1 (rows 16..31)
    const int rA0 = l16;
    const int rA1 = 16 + l16;
#pragma unroll 4
    for (int k0 = 0; k0 < 128; k0 += 4) {
      const int ka = k0 + 2 * half;         // A 16x4 layout: lanes16-31 hold K+2,K+3
      v2f a0 = *(const v2f*)&src[rA0 * 128 + ka];   // ds_load_b64
      v2f a1 = *(const v2f*)&src[rA1 * 128 + ka];
      v2f b;                                        // B 4x16 frag
      b.x = W[ka * 128 + n];
      b.y = W[(ka + 1) * 128 + n];
      c0 = __builtin_amdgcn_wmma_f32_16x16x4_f32(false, a0, false, b,
                                                 (short)0, c0, false, false);
      c1 = __builtin_amdgcn_wmma_f32_16x16x4_f32(false, a1, false, b,
                                                 (short)0, c1, false, false);
    }
#pragma unroll
    for (int g = 0; g < 8; ++g) {           // column-major scatter = permutation
      const int m = g + 8 * half;
      dst[n * 32 + m]      = c0[g];
      dst[n * 32 + 16 + m] = c1[g];
    }
  }
}

__global__ __launch_bounds__(NTHREADS, 1)
void monarch_block_linear_kernel(const float* __restrict__ x,
                                 const float* __restrict__ w1,
                                 const float* __restrict__ w2,
                                 float* __restrict__ out) {
  __shared__ float Ws1[128 * 128];      // 64 KB
  __shared__ float Ws2[128 * 128];      // 64 KB
  __shared__ float bufX[2][32 * 128];   // 2 x 16 KB: X ping-pong, Z' staging
  __shared__ float bufY[32 * 128];      // 16 KB: Y' (permuted intermediate)

  const int tid  = threadIdx.x;
  const int lane = tid & 31;
  const int wave = tid >> 5;
  const int half = lane >> 4;
  const int l16  = lane & 15;

  const long block_tok = (long)blockIdx.x * TPW;

#if USE_ASYNC
  async_copy_token(x + block_tok * 4096, bufX[0], tid);   // prologue: token 0
#endif

  // Cooperative weight load (once per block): 2 x 64 KB, b128 per lane.
#pragma unroll
  for (int i = tid * 4; i < 128 * 128; i += NTHREADS * 4) {
    *(v4f*)&Ws1[i] = *(const v4f*)&w1[i];
    *(v4f*)&Ws2[i] = *(const v4f*)&w2[i];
  }

  for (int t = 0; t < TPW; ++t) {
    const long token = block_tok + t;
    const int  cur   = t & 1;
    float* __restrict__ xb = bufX[cur];
    float* __restrict__ ot = out + token * 4096;

    // Protects bufX[1-cur] (prev token's store-phase ds reads) and bufY reuse;
    // on t==0 also covers the weight loads.
    __syncthreads();

#if USE_ASYNC
    if (t + 1 < TPW) {
      // Overlap: fetch next token into the other buffer while we compute.
      async_copy_token(x + (token + 1) * 4096, bufX[1 - cur], tid);
      WAIT_ASYNC(8);   // async loads retire in order: old 8 (token t) are done
    } else {
      WAIT_ASYNC(0);
    }
#else
    const float* xt = x + token * 4096;
#pragma unroll
    for (int i = tid * 4; i < 4096; i += NTHREADS * 4)
      *(v4f*)&xb[i] = *(const v4f*)&xt[i];
    if (t + 1 < TPW)
      __builtin_prefetch(xt + 4096 + tid * 32, 0, 0);
#endif

    __syncthreads();   // all waves' slice of X is in LDS

    // GEMM1 + permutation: Y' = perm(X @ W1)
    gemm_perm_step(xb, Ws1, bufY, wave, half, l16);
    __syncthreads();

    // GEMM2 + permutation: Z' = perm(Y' @ W2), into the consumed X buffer
    gemm_perm_step(bufY, Ws2, xb, wave, half, l16);
    __syncthreads();

    // Coalesced b128 store of the 4096-float result.
#pragma unroll
    for (int i = tid * 4; i < 4096; i += NTHREADS * 4)
      *(v4f*)&ot[i] = *(const v4f*)&xb[i];
  }
}

extern "C" void kernel_launch(void* const* d_in, const int* in_sizes, int n_in,
                              void* d_out, int out_size, void* d_ws, size_t ws_size,
                              hipStream_t stream) {
  const float* x  = (const float*)d_in[0];
  const float* w1 = (const float*)d_in[1];
  const float* w2 = (const float*)d_in[2];
  float* out      = (float*)d_out;

  const int tokens = in_sizes[0] / 4096;   // 32768
  const int blocks = tokens / TPW;         // 4096 workgroups

  hipLaunchKernelGGL(monarch_block_linear_kernel, dim3(blocks), dim3(NTHREADS),
                     0, stream, x, w1, w2, out);
}